// SparseMHA_20761871909447
// MI455X (gfx1250) — compile-verified
//
#include <hip/hip_runtime.h>
#include <hip/hip_bf16.h>
#include <math.h>

typedef __attribute__((ext_vector_type(2))) float v2f;
typedef __attribute__((ext_vector_type(8))) float v8f;

// One wave (32 lanes) per destination node. Edges of a node are contiguous
// because `row` is sorted. Flash-style online softmax over chunks of 16 edges.
// Per head, per chunk:
//   SDDMM: logits = Kgather(16e x 16d) * q_bcast(16d x 16n)   -> 4x chained v_wmma_f32_16x16x4_f32
//   SPMM : acc   += p_row0(16 x 16e)  * Vgather(16e x 16d)    -> 4x chained v_wmma_f32_16x16x4_f32 (C = running accumulator)
__global__ __launch_bounds__(256, 1)
void sparse_mha_wmma(const float* __restrict__ q,
                     const float* __restrict__ k,
                     const float* __restrict__ v,
                     const int*   __restrict__ row,
                     const int*   __restrict__ col,
                     float*       __restrict__ out,
                     int N, int E)
{
    const int wave = (int)(threadIdx.x >> 5);
    const int lane = (int)(threadIdx.x & 31);
    const int node = (int)blockIdx.x * 8 + wave;
    if (node >= N) return;                      // wave-uniform exit

    // ---- CSR range for this node: two binary searches on sorted `row` ----
    int lo = 0, hi = E;
    while (lo < hi) { int mid = (lo + hi) >> 1; if (row[mid] < node) lo = mid + 1; else hi = mid; }
    const int start = lo;
    hi = E;
    while (lo < hi) { int mid = (lo + hi) >> 1; if (row[mid] < node + 1) lo = mid + 1; else hi = mid; }
    const int end = lo;
    const int deg = end - start;

    if (deg == 0) {                             // empty segment -> zeros (wave-uniform)
        for (int t = lane; t < 128; t += 32) out[(size_t)node * 128 + t] = 0.0f;
        return;
    }

    const int  eLane  = lane & 15;              // edge slot this lane represents
    const bool hiHalf = (lane >= 16);
    const int  ofs    = hiHalf ? 2 : 0;         // K-slice offset for high half (WMMA A/B f32 layout)

    // q/k/v layout: [N, D=16, H=8] -> element (n,d,h) at n*128 + d*8 + h
    const float* qrow = q + (size_t)node * 128;

    for (int h = 0; h < 8; ++h) {
        float m = -INFINITY;                    // running max (replicated in all lanes)
        float s = 0.0f;                         // running sum of exp
        v8f acc = {0.f,0.f,0.f,0.f,0.f,0.f,0.f,0.f};   // SPMM accumulator (row 0 = live)

        for (int base = start; base < end; base += 16) {
            const int cnt  = min(16, end - base);
            const int eIdx = base + min(eLane, cnt - 1);   // clamped; masked later
            const int cSrc = col[eIdx];                    // col of edge `eLane`

            // ---------- SDDMM inputs ----------
            // A vgpr j (t=j>>1,b=j&1): lane<16 -> edge=lane,   K dim d = 4t+b
            //                          lane>=16 -> edge=lane-16, K dim d = 4t+b+2
            float a[8], b[8];
            const float* krow = k + (size_t)cSrc * 128 + h;
            #pragma unroll
            for (int j = 0; j < 8; ++j) {
                const int d = 4 * (j >> 1) + (j & 1) + ofs;
                a[j] = krow[d * 8];                        // k[cSrc, d, h]
                b[j] = qrow[d * 8 + h];                    // q[node, d, h] (uniform per half)
            }

            // ---------- SDDMM: 4 chained 16x16x4 f32 WMMAs ----------
            v8f dlog = {0.f,0.f,0.f,0.f,0.f,0.f,0.f,0.f};
            #pragma unroll
            for (int t = 0; t < 4; ++t) {
                v2f av = { a[2*t], a[2*t+1] };
                v2f bv = { b[2*t], b[2*t+1] };
                dlog = __builtin_amdgcn_wmma_f32_16x16x4_f32(
                           false, av, false, bv, (short)0, dlog, false, false);
            }
            // D[e,n] = logit[e] in every column n:
            //   lanes 0-15 hold e=0..7 in regs 0..7, lanes 16-31 hold e=8..15.

            // ---------- online softmax ----------
            float lg[8], lmax = -INFINITY;
            #pragma unroll
            for (int j = 0; j < 8; ++j) {
                const int e = j + (hiHalf ? 8 : 0);
                lg[j] = (e < cnt) ? dlog[j] : -INFINITY;   // mask padded edges
                lmax  = fmaxf(lmax, lg[j]);
            }
            lmax = fmaxf(lmax, __shfl_xor(lmax, 16, 32));  // combine halves
            const float mnew = fmaxf(m, lmax);
            const float corr = __expf(m - mnew);           // exp(-inf)=0 on first chunk
            m = mnew;

            float p[8], psum = 0.0f;
            #pragma unroll
            for (int j = 0; j < 8; ++j) {
                p[j]  = __expf(lg[j] - mnew);              // masked -> exp(-inf)=0
                psum += p[j];
            }
            psum += __shfl_xor(psum, 16, 32);
            s = s * corr + psum;

            #pragma unroll
            for (int j = 0; j < 8; ++j) acc[j] *= corr;    // rescale accumulator

            // ---------- SPMM A: row0 = p[e], all other rows 0 ----------
            // A vgpr j: lane0 needs p[4t+b], lane16 needs p[4t+2+b], others 0.
            float aa[8];
            #pragma unroll
            for (int j = 0; j < 8; ++j) {
                const int t = j >> 1, bsel = j & 1;
                const int e0 = 4 * t + bsel;               // wanted on lane 0
                const int e1 = 4 * t + 2 + bsel;           // wanted on lane 16
                const int i0 = (e0 < 8) ? e0 : (e0 - 8);
                const int i1 = (e1 >= 8) ? (e1 - 8) : e1;
                const float v0 = (e0 < 8)  ? p[i0] : __shfl_xor(p[i0], 16, 32);
                const float v1 = (e1 >= 8) ? p[i1] : __shfl_xor(p[i1], 16, 32);
                aa[j] = (lane == 0) ? v0 : ((lane == 16) ? v1 : 0.0f);
            }

            // ---------- SPMM B: Vgather, B[e, d] with d = lane%16 ----------
            float bb[8];
            #pragma unroll
            for (int j = 0; j < 8; ++j) {
                int e = 4 * (j >> 1) + (j & 1) + ofs;      // row (edge) for this lane half
                e = min(e, cnt - 1);                       // clamp; p=0 nulls padded rows
                const int ce = __shfl(cSrc, e, 32);        // col of edge e (bpermute)
                bb[j] = v[(size_t)ce * 128 + (size_t)(lane & 15) * 8 + h];
            }

            // ---------- SPMM: 4 chained WMMAs accumulating into acc ----------
            #pragma unroll
            for (int t = 0; t < 4; ++t) {
                v2f av = { aa[2*t], aa[2*t+1] };
                v2f bv = { bb[2*t], bb[2*t+1] };
                acc = __builtin_amdgcn_wmma_f32_16x16x4_f32(
                          false, av, false, bv, (short)0, acc, false, false);
            }
        } // chunk loop

        // Output: D row m=0 lives in acc[0] on lanes 0-15, column n = d = lane.
        const float o = acc[0] / s;
        if (lane < 16) out[(size_t)node * 128 + (size_t)lane * 8 + h] = o;
    } // head loop
}

extern "C" void kernel_launch(void* const* d_in, const int* in_sizes, int n_in,
                              void* d_out, int out_size, void* d_ws, size_t ws_size,
                              hipStream_t stream) {
    const float* q   = (const float*)d_in[0];   // [N, 16, 8]
    const float* k   = (const float*)d_in[1];   // [N, 16, 8]
    const float* v   = (const float*)d_in[2];   // [N, 16, 8]
    const int*   row = (const int*)  d_in[3];   // [E] sorted destination ids
    const int*   col = (const int*)  d_in[4];   // [E] source ids
    float*       out = (float*)d_out;           // [N, 128]

    const int N = in_sizes[0] / 128;
    const int E = in_sizes[3];

    const int waves_per_block = 8;              // 256 threads = 8 wave32
    dim3 grid((N + waves_per_block - 1) / waves_per_block);
    dim3 block(32 * waves_per_block);
    sparse_mha_wmma<<<grid, block, 0, stream>>>(q, k, v, row, col, out, N, E);
}